// CTCLossLayer_62491774157069
// MI455X (gfx1250) — compile-verified
//
#include <hip/hip_runtime.h>
#include <stdint.h>

// Problem constants (match reference).
#define Bb   1024
#define Tt   256
#define Cc   128
#define Ll   32
#define Ss   65          // 2*L + 1
#define PF   32          // async prefetch depth (ring slots); s_wait_asynccnt PF-1
#define NEGV (-1e30f)
#define EPSV (1e-7f)

// logaddexp in fp32: m + log(1 + exp(n - m)), n<=m so exp() in (0,1].
__device__ __forceinline__ float lae(float a, float b) {
    float m = fmaxf(a, b);
    float n = fminf(a, b);
    return m + __logf(1.0f + __expf(n - m));
}

// One async DMA: 16B per lane, global -> LDS (wave32: 512B row per issue).
__device__ __forceinline__ void async_row16(uint32_t lds_byte_addr, const float* gptr) {
    uint64_t ga = (uint64_t)(uintptr_t)gptr;
    asm volatile("global_load_async_to_lds_b128 %0, %1, off"
                 :: "v"(lds_byte_addr), "v"(ga) : "memory");
}

__device__ __forceinline__ void wait_async_pf() {
    asm volatile("s_wait_asynccnt 31" ::: "memory");   // keep PF-1 in flight
}

__global__ __launch_bounds__(32) void ctc_loss_kernel(
    const int*   __restrict__ y_true,        // (B, L)
    const float* __restrict__ y_pred,        // (B, T, C) softmax probs
    const int*   __restrict__ input_length,  // (B, 1)
    const int*   __restrict__ label_length,  // (B, 1)
    float*       __restrict__ out)           // (B, 1) = -log_lik
{
    __shared__ float ring[PF][Cc];           // 32 x 512B async-filled prob rows
    __shared__ float fin[Ss + 1];            // final alpha for readout

    const int b    = blockIdx.x;
    const int lane = threadIdx.x;            // 0..31, one wave per block

    // Lane i owns extended states s=2i (blank) and s=2i+1 (label i).
    // Lane 31 additionally owns s=64 (trailing blank).
    const int  lab  = y_true[b * Ll + lane];             // label for odd state
    const int  labp = __shfl_up(lab, 1, 32);             // label of lane-1
    const bool skip = (lane > 0) && (lab != labp);       // skip path for s=2i+1

    // --- async ring prefetch: rows 0..PF-1 --------------------------------
    const float*   base     = y_pred + (size_t)b * Tt * Cc;
    const uint32_t ringBase = (uint32_t)(uintptr_t)(&ring[0][0]);
    const uint32_t ldsLane  = (uint32_t)lane * 16u;
    #pragma unroll
    for (int k = 0; k < PF; ++k)
        async_row16(ringBase + (uint32_t)k * 512u + ldsLane, base + k * Cc + lane * 4);

    // --- alpha init (t = 0) -----------------------------------------------
    wait_async_pf();                                     // row 0 complete
    float lpb = __logf(ring[0][Cc - 1] + EPSV);          // log p(blank)
    float lpl = __logf(ring[0][lab]    + EPSV);          // log p(label_i)
    float ae  = (lane == 0) ? lpb : NEGV;                // alpha[2i],   s<=1 rule
    float ao  = (lane == 0) ? lpl : NEGV;                // alpha[2i+1]
    float a64 = NEGV;                                    // alpha[64] (lane 31)
    async_row16(ringBase + ldsLane, base + (size_t)PF * Cc + lane * 4); // refill slot 0

    const int inlen = input_length[b];

    // --- forward recursion over time --------------------------------------
    #pragma unroll 4
    for (int t = 1; t < Tt; ++t) {
        const int slot = t & (PF - 1);
        wait_async_pf();                                 // row t complete
        const float* r = ring[slot];

        lpb = __logf(r[Cc - 1] + EPSV);
        lpl = __logf(r[lab]    + EPSV);

        float pao = __shfl_up(ao, 1, 32);                // alpha[2i-1] (old)
        pao = (lane == 0) ? NEGV : pao;

        const float nae = lae(ae, pao) + lpb;            // even state: no skip
        const float a2  = skip ? pao : NEGV;
        const float nao = lae(lae(ao, ae), a2) + lpl;    // odd state
        const float n64 = lae(a64, ao) + lpb;            // s=64 (uses own ao=alpha[63])

        const bool upd = (t < inlen);
        ae  = upd ? nae : ae;
        ao  = upd ? nao : ao;
        a64 = upd ? n64 : a64;

        // Refill consumed slot with row t+PF (clamped duplicates at the tail
        // keep the outstanding-count invariant so the wait immediate is fixed).
        int nt = t + PF;
        if (nt > Tt - 1) nt = Tt - 1;
        async_row16(ringBase + (uint32_t)slot * 512u + ldsLane,
                    base + (size_t)nt * Cc + lane * 4);
    }

    // --- readout -----------------------------------------------------------
    fin[2 * lane]     = ae;
    fin[2 * lane + 1] = ao;
    if (lane == 31) fin[64] = a64;
    __syncthreads();                                     // single-wave: S_NOP + fence
    if (lane == 0) {
        const int   last = 2 * label_length[b];          // 2..64
        const float ll   = lae(fin[last], fin[last - 1]);
        out[b] = -ll;
    }
}

extern "C" void kernel_launch(void* const* d_in, const int* in_sizes, int n_in,
                              void* d_out, int out_size, void* d_ws, size_t ws_size,
                              hipStream_t stream) {
    (void)in_sizes; (void)n_in; (void)out_size; (void)d_ws; (void)ws_size;
    const int*   y_true = (const int*)d_in[0];
    const float* y_pred = (const float*)d_in[1];
    const int*   in_len = (const int*)d_in[2];
    const int*   lb_len = (const int*)d_in[3];
    float*       out    = (float*)d_out;
    hipLaunchKernelGGL(ctc_loss_kernel, dim3(Bb), dim3(32), 0, stream,
                       y_true, y_pred, in_len, lb_len, out);
}